// DQN_42803644072512
// MI455X (gfx1250) — compile-verified
//
#include <hip/hip_runtime.h>

typedef __attribute__((ext_vector_type(2))) float v2f;
typedef __attribute__((ext_vector_type(4))) float v4f;
typedef __attribute__((ext_vector_type(8))) float v8f;

#define NN 8192
#define DIM 64
#define T_ITERS 4

// ---------------------------------------------------------------------------
// Kernel 1: streaming scan of graph (256 MB) -> per-row s_pos / s_neg.
// Nontemporal B128 loads; this is the whole roofline (~11 us @ 23.3 TB/s).
// ---------------------------------------------------------------------------
__global__ void rowsum_kernel(const float* __restrict__ graph,
                              float* __restrict__ s_pos,
                              float* __restrict__ s_neg) {
  int row = blockIdx.x;
  const v4f* g = (const v4f*)(graph + (size_t)row * NN);
  float pos = 0.f, neg = 0.f;
#pragma unroll
  for (int k = 0; k < (NN / 4) / 256; ++k) {
    v4f val = __builtin_nontemporal_load(&g[threadIdx.x + 256 * k]);
    pos += fmaxf(val.x, 0.f) + fmaxf(val.y, 0.f) + fmaxf(val.z, 0.f) + fmaxf(val.w, 0.f);
    neg += fmaxf(-val.x, 0.f) + fmaxf(-val.y, 0.f) + fmaxf(-val.z, 0.f) + fmaxf(-val.w, 0.f);
  }
  __shared__ float rp[256], rn[256];
  int t = threadIdx.x;
  rp[t] = pos; rn[t] = neg;
  __syncthreads();
  for (int s = 128; s > 0; s >>= 1) {
    if (t < s) { rp[t] += rp[t + s]; rn[t] += rn[t + s]; }
    __syncthreads();
  }
  if (t == 0) { s_pos[row] = rp[0]; s_neg[row] = rn[0]; }
}

// ---------------------------------------------------------------------------
// Kernel 2: p = relu(theta4) @ theta3 ; q = relu(-theta4) @ theta3   (64 each)
// ---------------------------------------------------------------------------
__global__ void pq_kernel(const float* __restrict__ theta4,
                          const float* __restrict__ theta3,
                          float* __restrict__ p, float* __restrict__ q) {
  int j = threadIdx.x;
  float pj = 0.f, qj = 0.f;
  for (int k = 0; k < DIM; ++k) {
    float tv = theta4[k];
    float w  = theta3[k * DIM + j];
    pj += fmaxf(tv, 0.f) * w;
    qj += fmaxf(-tv, 0.f) * w;
  }
  p[j] = pj; q[j] = qj;
}

// ---------------------------------------------------------------------------
// Kernel 3: base[i][j] = x[i]*theta1[j] + s_pos[i]*p[j] + s_neg[i]*q[j]
// ---------------------------------------------------------------------------
__global__ void base_kernel(const int* __restrict__ x,
                            const float* __restrict__ theta1,
                            const float* __restrict__ s_pos,
                            const float* __restrict__ s_neg,
                            const float* __restrict__ p,
                            const float* __restrict__ q,
                            float* __restrict__ basep) {
  int idx = blockIdx.x * 256 + threadIdx.x;
  int i = idx >> 6, j = idx & 63;
  basep[idx] = (float)x[i] * theta1[j] + s_pos[i] * p[j] + s_neg[i] * q[j];
}

// ---------------------------------------------------------------------------
// Kernel 4: zero u (N x 64) and the column-sum accumulator S (64).
// ---------------------------------------------------------------------------
__global__ void zero_kernel(float* __restrict__ u, float* __restrict__ S) {
  int idx = blockIdx.x * 256 + threadIdx.x;
  u[idx] = 0.f;
  if (idx < DIM) S[idx] = 0.f;
}

// ---------------------------------------------------------------------------
// Kernel 5: Sth2 = S @ theta2 ; then reset S for next accumulation round.
// ---------------------------------------------------------------------------
__global__ void sth2_kernel(const float* __restrict__ theta2,
                            float* __restrict__ S,
                            float* __restrict__ sth2) {
  __shared__ float sS[DIM];
  int j = threadIdx.x;
  sS[j] = S[j];
  __syncthreads();
  float a = 0.f;
  for (int k = 0; k < DIM; ++k) a += sS[k] * theta2[k * DIM + j];
  sth2[j] = a;
  S[j] = 0.f;
}

// ---------------------------------------------------------------------------
// Kernel 6: u <- relu(base + Sth2 - u @ theta2), in place, and accumulate
// colsum(u_new) into S. GEMM done with V_WMMA_F32_16X16X4_F32 (full fp32).
// One wave owns a 16-row tile; 16 k-steps x 4 n-blocks of WMMA.
// A layout (16x4 f32): lane m=lane%16 is row M; kh=lane/16 selects K {0,1}
// vs {2,3}. B layout (4x16): lane m is column N; kh selects K pair.
// C/D: vgpr r = row (kh*8 + r), lane m = column.
// ---------------------------------------------------------------------------
__global__ void update_kernel(const float* __restrict__ basep,
                              const float* __restrict__ theta2,
                              const float* __restrict__ sth2,
                              float* __restrict__ u,
                              float* __restrict__ S) {
  __shared__ float t2[DIM * DIM];   // 16 KB theta2 tile
  __shared__ float ssth[DIM];
  __shared__ float csum[DIM];
  int t = threadIdx.x;

  { // cooperative load of theta2 into LDS (B128 per thread x4)
    const v4f* src = (const v4f*)theta2;
    v4f* dst = (v4f*)t2;
#pragma unroll
    for (int k = 0; k < (DIM * DIM / 4) / 256; ++k)
      dst[t + 256 * k] = src[t + 256 * k];
  }
  if (t < DIM) { ssth[t] = sth2[t]; csum[t] = 0.f; }
  __syncthreads();

  int wave = t >> 5, lane = t & 31;
  int m = lane & 15, kh = lane >> 4;
  int row0 = (blockIdx.x * 8 + wave) * 16;

  // A fragments: this wave's 16 rows of u, fully read before any write (in-place safe)
  const float* urow = u + (size_t)(row0 + m) * DIM + 2 * kh;
  v2f afr[16];
#pragma unroll
  for (int kb = 0; kb < 16; ++kb) {
    afr[kb].x = urow[4 * kb];
    afr[kb].y = urow[4 * kb + 1];
  }

#pragma unroll
  for (int nb = 0; nb < 4; ++nb) {
    v8f acc = {0.f, 0.f, 0.f, 0.f, 0.f, 0.f, 0.f, 0.f};
    int col = nb * 16 + m;
#pragma unroll
    for (int kb = 0; kb < 16; ++kb) {
      v2f bfr;
      bfr.x = t2[(4 * kb + 2 * kh) * DIM + col];
      bfr.y = t2[(4 * kb + 2 * kh + 1) * DIM + col];
      acc = __builtin_amdgcn_wmma_f32_16x16x4_f32(
          /*neg_a=*/false, afr[kb], /*neg_b=*/false, bfr,
          /*c_mod=*/(short)0, acc, /*reuse_a=*/false, /*reuse_b=*/false);
    }
    float sv = ssth[col];
    float cs = 0.f;
#pragma unroll
    for (int r = 0; r < 8; ++r) {
      int row = row0 + kh * 8 + r;
      float d = basep[(size_t)row * DIM + col] + sv - acc[r];
      d = fmaxf(d, 0.f);
      u[(size_t)row * DIM + col] = d;
      cs += d;
    }
    atomicAdd(&csum[col], cs);   // ds_add_f32, block-local reduce
  }
  __syncthreads();
  if (t < DIM) atomicAdd(&S[t], csum[t]);  // 64 global atomics per block
}

// ---------------------------------------------------------------------------
// Kernel 7: out = relu(concat(S@theta6, u[v]@theta7)) @ theta5  (scalar)
// ---------------------------------------------------------------------------
__global__ void final_kernel(const float* __restrict__ S,
                             const float* __restrict__ u,
                             const float* __restrict__ theta6,
                             const float* __restrict__ theta7,
                             const float* __restrict__ theta5,
                             const int* __restrict__ vptr,
                             float* __restrict__ out) {
  __shared__ float sS[DIM], sU[DIM], red[128];
  int t = threadIdx.x;
  if (t < DIM) {
    int v = *vptr;
    sS[t] = S[t];
    sU[t] = u[(size_t)v * DIM + t];
  }
  __syncthreads();
  float r;
  if (t < DIM) {
    float a = 0.f;
    for (int k = 0; k < DIM; ++k) a += sS[k] * theta6[k * DIM + t];
    r = fmaxf(a, 0.f) * theta5[t];
  } else {
    int j = t - DIM;
    float a = 0.f;
    for (int k = 0; k < DIM; ++k) a += sU[k] * theta7[k * DIM + j];
    r = fmaxf(a, 0.f) * theta5[DIM + j];
  }
  red[t] = r;
  __syncthreads();
  for (int s = 64; s > 0; s >>= 1) {
    if (t < s) red[t] += red[t + s];
    __syncthreads();
  }
  if (t == 0) out[0] = red[0];
}

// ---------------------------------------------------------------------------
extern "C" void kernel_launch(void* const* d_in, const int* in_sizes, int n_in,
                              void* d_out, int out_size, void* d_ws, size_t ws_size,
                              hipStream_t stream) {
  const float* graph  = (const float*)d_in[0];
  const int*   x      = (const int*)d_in[1];
  const float* theta1 = (const float*)d_in[2];
  const float* theta2 = (const float*)d_in[3];
  const float* theta3 = (const float*)d_in[4];
  const float* theta4 = (const float*)d_in[5];
  const float* theta5 = (const float*)d_in[6];
  const float* theta6 = (const float*)d_in[7];
  const float* theta7 = (const float*)d_in[8];
  const int*   vptr   = (const int*)d_in[9];
  float* out = (float*)d_out;

  float* ws    = (float*)d_ws;
  float* s_pos = ws;               ws += NN;
  float* s_neg = ws;               ws += NN;
  float* p     = ws;               ws += DIM;
  float* q     = ws;               ws += DIM;
  float* basep = ws;               ws += (size_t)NN * DIM;
  float* u     = ws;               ws += (size_t)NN * DIM;
  float* S     = ws;               ws += DIM;
  float* sth2  = ws;               ws += DIM;

  rowsum_kernel<<<NN, 256, 0, stream>>>(graph, s_pos, s_neg);
  pq_kernel<<<1, DIM, 0, stream>>>(theta4, theta3, p, q);
  base_kernel<<<(NN * DIM) / 256, 256, 0, stream>>>(x, theta1, s_pos, s_neg, p, q, basep);
  zero_kernel<<<(NN * DIM) / 256, 256, 0, stream>>>(u, S);

  for (int it = 0; it < T_ITERS; ++it) {
    sth2_kernel<<<1, DIM, 0, stream>>>(theta2, S, sth2);
    update_kernel<<<NN / (16 * 8), 256, 0, stream>>>(basep, theta2, sth2, u, S);
  }

  final_kernel<<<1, 128, 0, stream>>>(S, u, theta6, theta7, theta5, vptr, out);
}